// GCNNet_23519240913379
// MI455X (gfx1250) — compile-verified
//
#include <hip/hip_runtime.h>
#include <hip/hip_bf16.h>

typedef __attribute__((ext_vector_type(2))) float v2f;
typedef __attribute__((ext_vector_type(8))) float v8f;

#define D_FEAT 64
#define N_GRAPHS_C 64

// ---------------------------------------------------------------------------
// Degree / normalization: deg starts at 1.0 (self-loop), edges atomically add,
// then transform in place to deg^{-1/2}.
// ---------------------------------------------------------------------------
__global__ __launch_bounds__(256) void deg_init_kernel(float* __restrict__ deg, int n) {
    int i = blockIdx.x * blockDim.x + threadIdx.x;
    if (i < n) deg[i] = 1.0f;  // self-loop contribution
}

__global__ __launch_bounds__(256) void deg_count_kernel(const int* __restrict__ col,
                                                        float* __restrict__ deg, int e) {
    int i = blockIdx.x * blockDim.x + threadIdx.x;
    if (i < e) atomicAdd(&deg[col[i]], 1.0f);
}

__global__ __launch_bounds__(256) void deg_finish_kernel(float* __restrict__ deg, int n) {
    int i = blockIdx.x * blockDim.x + threadIdx.x;
    if (i < n) {
        float d = deg[i];
        deg[i] = (d > 0.0f) ? rsqrtf(d) : 0.0f;
    }
}

// ---------------------------------------------------------------------------
// H[nrows x 64] = X[nrows x 64] @ W[64 x 64] using V_WMMA_F32_16X16X4_F32.
// One wave computes a full 16x64 strip: 4 accumulator tiles share each A
// fragment (4x reuse of the big operand), 64 WMMAs per wave, 4 independent
// accumulation chains per k-step. nrows = 50000 = 3125*16; whole-wave exit
// keeps EXEC all-ones for every executed WMMA.
//
// 32-bit fragment layouts (ISA 7.12.2):
//   A 16x4:  lane<16 -> K=k,k+1 in the 2 VGPRs; lane>=16 -> K=k+2,k+3
//   B 4x16:  mirrored (lane&15 = N, half-wave selects K pair)
//   C/D:     VGPR r -> M = r (lanes 0-15) / r+8 (lanes 16-31), N = lane&15
// ---------------------------------------------------------------------------
__global__ __launch_bounds__(256) void gemm_wmma_kernel(const float* __restrict__ X,
                                                        const float* __restrict__ W,
                                                        float* __restrict__ H,
                                                        int nrows) {
    const int wave = (int)((blockIdx.x * blockDim.x + threadIdx.x) >> 5);
    const int lane = (int)(threadIdx.x & 31);
    const int row0 = wave * 16;
    if (row0 >= nrows) return;              // wave-uniform exit
    const int lm = lane & 15;               // M index for A, N index for B/D
    const int kk = (lane >> 4) * 2;         // half-wave K sub-offset (0 or 2)

    const float* __restrict__ xrow = X + (size_t)(row0 + lm) * D_FEAT + kk;

    v8f c0 = {}, c1 = {}, c2 = {}, c3 = {};
#pragma unroll
    for (int k = 0; k < D_FEAT; k += 4) {
        v2f a;
        a.x = xrow[k + 0];                              // A: K = k+kk
        a.y = xrow[k + 1];                              // A: K = k+kk+1
        const float* __restrict__ wr0 = W + (size_t)(k + kk) * D_FEAT + lm;
        const float* __restrict__ wr1 = wr0 + D_FEAT;
        v2f b0, b1, b2, b3;
        b0.x = wr0[0];   b0.y = wr1[0];                 // n-tile 0
        b1.x = wr0[16];  b1.y = wr1[16];                // n-tile 1
        b2.x = wr0[32];  b2.y = wr1[32];                // n-tile 2
        b3.x = wr0[48];  b3.y = wr1[48];                // n-tile 3
        c0 = __builtin_amdgcn_wmma_f32_16x16x4_f32(false, a, false, b0, (short)0, c0, false, false);
        c1 = __builtin_amdgcn_wmma_f32_16x16x4_f32(false, a, false, b1, (short)0, c1, false, false);
        c2 = __builtin_amdgcn_wmma_f32_16x16x4_f32(false, a, false, b2, (short)0, c2, false, false);
        c3 = __builtin_amdgcn_wmma_f32_16x16x4_f32(false, a, false, b3, (short)0, c3, false, false);
    }

    const int mbase = row0 + (lane >> 4) * 8;
#pragma unroll
    for (int r = 0; r < 8; ++r) {
        float* __restrict__ hr = H + (size_t)(mbase + r) * D_FEAT + lm;
        hr[0]  = c0[r];
        hr[16] = c1[r];
        hr[32] = c2[r];
        hr[48] = c3[r];
    }
}

// ---------------------------------------------------------------------------
// ACC[i][d] = bias[d] + H[i][d] * dinv[i]^2   (self-loop message folded in;
// also serves as the zero-initialization of the segment-sum accumulator)
// ---------------------------------------------------------------------------
__global__ __launch_bounds__(256) void init_acc_kernel(const float* __restrict__ H,
                                                       const float* __restrict__ dinv,
                                                       const float* __restrict__ bias,
                                                       float* __restrict__ ACC, int n) {
    int t = blockIdx.x * blockDim.x + threadIdx.x;
    if (t >= n) return;
    int node = t >> 6;
    int d    = t & 63;
    float di = dinv[node];
    ACC[t] = bias[d] + H[t] * di * di;
}

// ---------------------------------------------------------------------------
// Edge scatter: 16 threads per edge, each handles a float4 chunk of the
// 64-wide feature row. msg = H[row] * dinv[row]*dinv[col]; atomic-add to
// ACC[col]. Working set (~35 MB) is resident in the 192 MB L2, so the
// gathers and fp32 atomics resolve at L2 bandwidth.
// ---------------------------------------------------------------------------
__global__ __launch_bounds__(256) void scatter_kernel(const float* __restrict__ H,
                                                      const int* __restrict__ row,
                                                      const int* __restrict__ col,
                                                      const float* __restrict__ dinv,
                                                      float* __restrict__ ACC, int e) {
    long long t = (long long)blockIdx.x * blockDim.x + threadIdx.x;
    int edge = (int)(t >> 4);
    if (edge >= e) return;
    int f = (int)(t & 15) * 4;
    int r = row[edge];
    int c = col[edge];
    float s = dinv[r] * dinv[c];
    const float4 hv = *(const float4*)(H + (size_t)r * D_FEAT + f);
    float* dst = ACC + (size_t)c * D_FEAT + f;
    atomicAdd(dst + 0, hv.x * s);
    atomicAdd(dst + 1, hv.y * s);
    atomicAdd(dst + 2, hv.z * s);
    atomicAdd(dst + 3, hv.w * s);
}

__global__ __launch_bounds__(256) void relu_kernel(float* __restrict__ A, int n) {
    int i = blockIdx.x * blockDim.x + threadIdx.x;
    if (i < n) A[i] = fmaxf(A[i], 0.0f);
}

// ---------------------------------------------------------------------------
// Global mean pool
// ---------------------------------------------------------------------------
__global__ __launch_bounds__(256) void pool_zero_kernel(float* __restrict__ pool,
                                                        float* __restrict__ cnt, int n) {
    int i = blockIdx.x * blockDim.x + threadIdx.x;
    if (i < n) pool[i] = 0.0f;
    if (i < N_GRAPHS_C) cnt[i] = 0.0f;
}

__global__ __launch_bounds__(256) void pool_kernel(const float* __restrict__ H,
                                                   const int* __restrict__ batch,
                                                   float* __restrict__ pool,
                                                   float* __restrict__ cnt, int nnodes) {
    long long t = (long long)blockIdx.x * blockDim.x + threadIdx.x;
    int node = (int)(t >> 4);
    if (node >= nnodes) return;
    int f = (int)(t & 15) * 4;
    int g = batch[node];
    const float4 hv = *(const float4*)(H + (size_t)node * D_FEAT + f);
    float* dst = pool + (size_t)g * D_FEAT + f;
    atomicAdd(dst + 0, hv.x);
    atomicAdd(dst + 1, hv.y);
    atomicAdd(dst + 2, hv.z);
    atomicAdd(dst + 3, hv.w);
    if (f == 0) atomicAdd(&cnt[g], 1.0f);
}

__global__ __launch_bounds__(256) void pool_final_kernel(const float* __restrict__ pool,
                                                         const float* __restrict__ cnt,
                                                         float* __restrict__ out, int n) {
    int i = blockIdx.x * blockDim.x + threadIdx.x;
    if (i < n) {
        int g = i >> 6;
        out[i] = pool[i] / fmaxf(cnt[g], 1.0f);
    }
}

// ---------------------------------------------------------------------------
// Launcher
// ---------------------------------------------------------------------------
extern "C" void kernel_launch(void* const* d_in, const int* in_sizes, int n_in,
                              void* d_out, int out_size, void* d_ws, size_t ws_size,
                              hipStream_t stream) {
    const float* x     = (const float*)d_in[0];
    const int*   ei    = (const int*)d_in[1];   // [2, E]
    const int*   batch = (const int*)d_in[2];
    const float* W1    = (const float*)d_in[3];
    const float* b1    = (const float*)d_in[4];
    const float* W2    = (const float*)d_in[5];
    const float* b2    = (const float*)d_in[6];
    const float* W3    = (const float*)d_in[7];
    const float* b3    = (const float*)d_in[8];
    float* out = (float*)d_out;

    const int N = in_sizes[0] / D_FEAT;   // 50000
    const int E = in_sizes[1] / 2;        // 800000
    const int* erow = ei;
    const int* ecol = ei + E;

    // Workspace layout (floats)
    float* w = (float*)d_ws;
    size_t off = 0;
    float* dinv = w + off; off += (size_t)((N + 63) / 64) * 64;
    float* bufA = w + off; off += (size_t)N * D_FEAT;   // h = X @ W (gather src)
    float* bufB = w + off; off += (size_t)N * D_FEAT;   // accumulator / next input
    float* pool = w + off; off += (size_t)N_GRAPHS_C * D_FEAT;
    float* cnt  = w + off; off += N_GRAPHS_C;

    const int ND = N * D_FEAT;
    const int blk = 256;

    // --- normalization ---
    deg_init_kernel<<<(N + blk - 1) / blk, blk, 0, stream>>>(dinv, N);
    deg_count_kernel<<<(E + blk - 1) / blk, blk, 0, stream>>>(ecol, dinv, E);
    deg_finish_kernel<<<(N + blk - 1) / blk, blk, 0, stream>>>(dinv, N);

    // GEMM grid: one wave per 16-row strip -> N/16 waves
    const int waves = (N + 15) / 16;
    const int gemm_blocks = (waves * 32 + blk - 1) / blk;
    const long long sthreads = (long long)E * 16;
    const int scat_blocks = (int)((sthreads + blk - 1) / blk);
    const int nd_blocks = (ND + blk - 1) / blk;

    // --- layer 1 ---
    gemm_wmma_kernel<<<gemm_blocks, blk, 0, stream>>>(x, W1, bufA, N);
    init_acc_kernel<<<nd_blocks, blk, 0, stream>>>(bufA, dinv, b1, bufB, ND);
    scatter_kernel<<<scat_blocks, blk, 0, stream>>>(bufA, erow, ecol, dinv, bufB, E);
    relu_kernel<<<nd_blocks, blk, 0, stream>>>(bufB, ND);

    // --- layer 2 ---
    gemm_wmma_kernel<<<gemm_blocks, blk, 0, stream>>>(bufB, W2, bufA, N);
    init_acc_kernel<<<nd_blocks, blk, 0, stream>>>(bufA, dinv, b2, bufB, ND);
    scatter_kernel<<<scat_blocks, blk, 0, stream>>>(bufA, erow, ecol, dinv, bufB, E);
    relu_kernel<<<nd_blocks, blk, 0, stream>>>(bufB, ND);

    // --- layer 3 ---
    gemm_wmma_kernel<<<gemm_blocks, blk, 0, stream>>>(bufB, W3, bufA, N);
    init_acc_kernel<<<nd_blocks, blk, 0, stream>>>(bufA, dinv, b3, bufB, ND);
    scatter_kernel<<<scat_blocks, blk, 0, stream>>>(bufA, erow, ecol, dinv, bufB, E);
    relu_kernel<<<nd_blocks, blk, 0, stream>>>(bufB, ND);

    // --- global mean pool ---
    pool_zero_kernel<<<(N_GRAPHS_C * D_FEAT + blk - 1) / blk, blk, 0, stream>>>(pool, cnt,
                                                                               N_GRAPHS_C * D_FEAT);
    const long long pthreads = (long long)N * 16;
    pool_kernel<<<(int)((pthreads + blk - 1) / blk), blk, 0, stream>>>(bufB, batch, pool, cnt, N);
    pool_final_kernel<<<(N_GRAPHS_C * D_FEAT + blk - 1) / blk, blk, 0, stream>>>(pool, cnt, out,
                                                                                 N_GRAPHS_C * D_FEAT);
}